// MambaSSM_38646115730085
// MI455X (gfx1250) — compile-verified
//
#include <hip/hip_runtime.h>
#include <hip/hip_bf16.h>
#include <math.h>

// ---------------- problem constants ----------------
#define D_MODEL  511
#define D_INNER  1022
#define D_STATE  16
#define D_CONV   4
#define DT_RANK  32
#define SEQ      2048
#define N_LAYERS 4

// padded leading dims
#define LD_X     512     // X: SEQ x 512   (cols 0..510 valid)
#define LD_XR    2048    // XR: SEQ x 2048 (cols 0..2043 valid; xin=0..1021, res=1022..2043)
#define LD_XC    1024    // XC: SEQ x 1024 (cols 0..1021 valid)
#define LD_XDBL  64      // XDBL: SEQ x 64 (dt 0..31, B 32..47, C 48..63)
#define LD_DELTA 1024
#define LD_YG    1024

#define LDS_STRIDE 36    // dwords; keeps b128 staging stores 16B-aligned,
                         // b64 fragment reads 8B-aligned, banks conflict-free

typedef float v2f __attribute__((ext_vector_type(2)));
typedef float v4f __attribute__((ext_vector_type(4)));
typedef float v8f __attribute__((ext_vector_type(8)));

__device__ __forceinline__ float silu_f(float x) {
    return x / (1.0f + __expf(-x));
}

// =====================================================================
// WMMA f32 GEMM:  C[m,n] (+)= sum_k A[m,k] * B[n,k]     (B row-major NxK)
// EPI: 0 = store, 1 = softplus(acc + bias[n]) store, 2 = C += acc
// Block tile 64x64, 8 waves of 32 lanes, each wave 32(M)x16(N).
// A is always a padded workspace buffer => lda % 4 == 0 (16B rows).
// =====================================================================
template<int EPI>
__global__ __launch_bounds__(256)
void wmma_gemm_f32(const float* __restrict__ A, int lda,
                   const float* __restrict__ B, int ldb,
                   float* __restrict__ C, int ldc,
                   int Nact, int Kact,
                   const float* __restrict__ bias)
{
    __shared__ alignas(16) float As[64 * LDS_STRIDE];
    __shared__ alignas(16) float Bs[64 * LDS_STRIDE];

    const int tid   = threadIdx.x;
    const int lane  = tid & 31;
    const int wave  = tid >> 5;
    const int wm    = (wave & 1) * 32;        // wave M offset: 0 / 32
    const int wn    = (wave >> 1) * 16;       // wave N offset: 0/16/32/48
    const int mBase = blockIdx.y * 64;
    const int nBase = blockIdx.x * 64;
    const int l16   = lane & 15;
    const int half  = lane >> 4;              // 0 or 1
    const int halfk = half * 2;               // K sub-offset per ISA layout

    const bool fullN = (nBase + 64) <= Nact;

    v8f acc0 = {};
    v8f acc1 = {};

    const int ktiles = (Kact + 31) >> 5;
    for (int kt = 0; kt < ktiles; ++kt) {
        const int k0 = kt << 5;
        const bool fullK = (k0 + 32) <= Kact;

        if (fullK && fullN) {
            // -------- fast path: no guards --------
            // A: two b128 loads/stores per thread (rows 16B aligned)
            const int am = tid >> 3;                 // 0..31
            const int ak = (tid & 7) << 2;           // 0,4,...,28
            #pragma unroll
            for (int p = 0; p < 2; ++p) {
                int m = am + p * 32;
                v4f av = *(const v4f*)&A[(size_t)(mBase + m) * lda + (k0 + ak)];
                *(v4f*)&As[m * LDS_STRIDE + ak] = av;
            }
            // B: 8 unguarded scalar loads (rows only 4B aligned in general)
            #pragma unroll
            for (int i = 0; i < 8; ++i) {
                int idx = tid + (i << 8);
                int m   = idx >> 5;
                int k   = idx & 31;
                Bs[m * LDS_STRIDE + k] = B[(size_t)(nBase + m) * ldb + (k0 + k)];
            }
            // prefetch next K tile of A into cache
            if (kt + 1 < ktiles) {
                __builtin_prefetch(&A[(size_t)(mBase + am) * lda + (k0 + 32 + ak)], 0, 1);
            }
        } else {
            // -------- guarded path (partial tiles only) --------
            #pragma unroll
            for (int i = 0; i < 8; ++i) {
                int idx = tid + (i << 8);
                int m   = idx >> 5;
                int k   = idx & 31;
                bool kok = (k0 + k) < Kact;
                float av = kok ? A[(size_t)(mBase + m) * lda + (k0 + k)] : 0.0f;
                float bv = (kok && (nBase + m) < Nact)
                             ? B[(size_t)(nBase + m) * ldb + (k0 + k)] : 0.0f;
                As[m * LDS_STRIDE + k] = av;
                Bs[m * LDS_STRIDE + k] = bv;
            }
        }
        __syncthreads();

        // ---- 8 x K4 WMMA steps
        #pragma unroll
        for (int kk = 0; kk < 32; kk += 4) {
            v2f a0 = *(const v2f*)&As[(wm +      l16) * LDS_STRIDE + kk + halfk];
            v2f a1 = *(const v2f*)&As[(wm + 16 + l16) * LDS_STRIDE + kk + halfk];
            v2f b  = *(const v2f*)&Bs[(wn +      l16) * LDS_STRIDE + kk + halfk];
            acc0 = __builtin_amdgcn_wmma_f32_16x16x4_f32(
                       false, a0, false, b, (short)0, acc0, false, false);
            acc1 = __builtin_amdgcn_wmma_f32_16x16x4_f32(
                       false, a1, false, b, (short)0, acc1, false, false);
        }
        __syncthreads();
    }

    // ---- epilogue ----
    const int n = nBase + wn + l16;
    if (n >= Nact) return;
    float bval = (EPI == 1) ? bias[n] : 0.0f;

    #pragma unroll
    for (int r = 0; r < 8; ++r) {
        int ml  = r + (half << 3);            // local row per C/D layout
        int mg0 = mBase + wm + ml;            // subtile 0 row
        int mg1 = mg0 + 16;                   // subtile 1 row
        float v0 = acc0[r];
        float v1 = acc1[r];
        if (EPI == 0) {
            C[(size_t)mg0 * ldc + n] = v0;
            C[(size_t)mg1 * ldc + n] = v1;
        } else if (EPI == 1) {
            float s0 = v0 + bval, s1 = v1 + bval;
            s0 = (s0 > 20.0f) ? s0 : log1pf(__expf(s0));
            s1 = (s1 > 20.0f) ? s1 : log1pf(__expf(s1));
            C[(size_t)mg0 * ldc + n] = s0;
            C[(size_t)mg1 * ldc + n] = s1;
        } else {
            C[(size_t)mg0 * ldc + n] += v0;
            C[(size_t)mg1 * ldc + n] += v1;
        }
    }
}

// =====================================================================
// Depthwise causal conv (width 4) + bias + SiLU.  XR xin cols -> XC.
// =====================================================================
__global__ __launch_bounds__(256)
void conv_silu_kernel(const float* __restrict__ XR,
                      const float* __restrict__ cw,
                      const float* __restrict__ cb,
                      float* __restrict__ XC)
{
    int c = blockIdx.x * 256 + threadIdx.x;   // 0..1023
    int s = blockIdx.y;                       // 0..2047
    if (c >= LD_XC) return;
    float v = 0.0f;
    if (c < D_INNER) {
        float acc = cb[c];
        #pragma unroll
        for (int j = 0; j < D_CONV; ++j) {
            int ss = s - (D_CONV - 1) + j;
            if (ss >= 0) acc += cw[c * D_CONV + j] * XR[(size_t)ss * LD_XR + c];
        }
        v = silu_f(acc);
    }
    XC[(size_t)s * LD_XC + c] = v;
}

// =====================================================================
// Selective scan. One thread per (d, n) state; n = lane%16 so the
// y = sum_n h*C reduction is 4 shfl_xor steps inside a wave32 half.
// Unrolled x4: batch the (h-independent) operand loads for 4 timesteps,
// then run 4 dependent recurrence steps -> 4x memory-latency overlap
// per serial-FMA chain link.
// Fused: +u*D and *silu(res), writes gated y for the out_proj GEMM.
// =====================================================================
__global__ __launch_bounds__(256)
void scan_kernel(const float* __restrict__ DELTA,
                 const float* __restrict__ XC,
                 const float* __restrict__ XDBL,
                 const float* __restrict__ XR,     // res at cols 1022..2043
                 const float* __restrict__ A_log,
                 const float* __restrict__ Dvec,
                 float* __restrict__ YG)
{
    int p = blockIdx.x * 256 + threadIdx.x;
    int d = p >> 4;
    int n = p & 15;
    bool active = (d < D_INNER);
    int dc = active ? d : (D_INNER - 1);      // clamped for uniform loads

    float a  = -__expf(A_log[dc * D_STATE + n]);
    float Dd = Dvec[dc];
    float h  = 0.0f;

    for (int t0 = 0; t0 < SEQ; t0 += 4) {
        float delta[4], u[4], Bn[4], Cn[4], res[4];
        #pragma unroll
        for (int j = 0; j < 4; ++j) {
            size_t t = t0 + j;
            delta[j] = DELTA[t * LD_DELTA + dc];
            u[j]     = XC[t * LD_XC + dc];
            Bn[j]    = XDBL[t * LD_XDBL + DT_RANK + n];
            Cn[j]    = XDBL[t * LD_XDBL + DT_RANK + D_STATE + n];
            res[j]   = XR[t * LD_XR + D_INNER + dc];
        }
        #pragma unroll
        for (int j = 0; j < 4; ++j) {
            float dA = __expf(delta[j] * a);
            h = dA * h + delta[j] * Bn[j] * u[j];

            float y = h * Cn[j];
            y += __shfl_xor(y, 8);
            y += __shfl_xor(y, 4);
            y += __shfl_xor(y, 2);
            y += __shfl_xor(y, 1);

            if (active && n == 0) {
                YG[(size_t)(t0 + j) * LD_YG + d] =
                    (y + u[j] * Dd) * silu_f(res[j]);
            }
        }
    }
}

// =====================================================================
// Transposes in/out
// =====================================================================
__global__ void transpose_in_kernel(const float* __restrict__ Z,
                                    float* __restrict__ X)
{
    int s = blockIdx.x * 256 + threadIdx.x;
    int d = blockIdx.y;
    if (s < SEQ) X[(size_t)s * LD_X + d] = Z[(size_t)d * SEQ + s];
}

__global__ void transpose_out_kernel(const float* __restrict__ X,
                                     float* __restrict__ OUT)
{
    int s = blockIdx.x * 256 + threadIdx.x;
    int d = blockIdx.y;
    if (s < SEQ) OUT[(size_t)d * SEQ + s] = X[(size_t)s * LD_X + d];
}

// =====================================================================
extern "C" void kernel_launch(void* const* d_in, const int* in_sizes, int n_in,
                              void* d_out, int out_size, void* d_ws, size_t ws_size,
                              hipStream_t stream)
{
    const float* Z         = (const float*)d_in[0];
    const float* in_proj_w = (const float*)d_in[1];   // (2044, 511)
    const float* conv_w    = (const float*)d_in[2];   // (1022, 1, 4)
    const float* conv_b    = (const float*)d_in[3];   // (1022)
    const float* x_proj_w  = (const float*)d_in[4];   // (64, 1022)
    const float* dt_proj_w = (const float*)d_in[5];   // (1022, 32)
    const float* dt_proj_b = (const float*)d_in[6];   // (1022)
    const float* A_log     = (const float*)d_in[7];   // (1022, 16)
    const float* Dvec      = (const float*)d_in[8];   // (1022)
    const float* out_proj_w= (const float*)d_in[9];   // (511, 1022)

    // ---- workspace carve (floats) ----
    float* base  = (float*)d_ws;
    float* X     = base;                               // 2048*512
    float* XR    = X     + (size_t)SEQ * LD_X;         // 2048*2048
    float* XC    = XR    + (size_t)SEQ * LD_XR;        // 2048*1024
    float* XDBL  = XC    + (size_t)SEQ * LD_XC;        // 2048*64
    float* DELTA = XDBL  + (size_t)SEQ * LD_XDBL;      // 2048*1024
    float* YG    = DELTA + (size_t)SEQ * LD_DELTA;     // 2048*1024

    dim3 blk(256);

    // x = Z^T
    transpose_in_kernel<<<dim3(SEQ / 256, D_MODEL), blk, 0, stream>>>(Z, X);

    for (int layer = 0; layer < N_LAYERS; ++layer) {
        // 1) XR = X @ in_proj_w^T           M=2048 N=2044 K=511
        wmma_gemm_f32<0><<<dim3((2 * D_INNER + 63) / 64, SEQ / 64), blk, 0, stream>>>(
            X, LD_X, in_proj_w, D_MODEL, XR, LD_XR, 2 * D_INNER, D_MODEL, nullptr);

        // 2) XC = silu(depthwise_conv(XR[:, :1022]) + b)
        conv_silu_kernel<<<dim3(LD_XC / 256, SEQ), blk, 0, stream>>>(XR, conv_w, conv_b, XC);

        // 3) XDBL = XC @ x_proj_w^T         M=2048 N=64 K=1022
        wmma_gemm_f32<0><<<dim3(1, SEQ / 64), blk, 0, stream>>>(
            XC, LD_XC, x_proj_w, D_INNER, XDBL, LD_XDBL, DT_RANK + 2 * D_STATE, D_INNER, nullptr);

        // 4) DELTA = softplus(dt @ dt_proj_w^T + b)   M=2048 N=1022 K=32
        wmma_gemm_f32<1><<<dim3((D_INNER + 63) / 64, SEQ / 64), blk, 0, stream>>>(
            XDBL, LD_XDBL, dt_proj_w, DT_RANK, DELTA, LD_DELTA, D_INNER, DT_RANK, dt_proj_b);

        // 5) selective scan + gating -> YG
        scan_kernel<<<dim3((D_INNER * D_STATE + 255) / 256), blk, 0, stream>>>(
            DELTA, XC, XDBL, XR, A_log, Dvec, YG);

        // 6) X += YG @ out_proj_w^T (residual)  M=2048 N=511 K=1022
        wmma_gemm_f32<2><<<dim3((D_MODEL + 63) / 64, SEQ / 64), blk, 0, stream>>>(
            YG, LD_YG, out_proj_w, D_INNER, X, LD_X, D_MODEL, D_INNER, nullptr);
    }

    // out = X^T
    transpose_out_kernel<<<dim3(SEQ / 256, D_MODEL), blk, 0, stream>>>(X, (float*)d_out);
}